// Model_41875931136506
// MI455X (gfx1250) — compile-verified
//
#include <hip/hip_runtime.h>
#include <hip/hip_bf16.h>

// Problem sizes (fixed by the reference)
#define BB 16
#define TT 2048
#define FF 512

typedef float v2f __attribute__((ext_vector_type(2)));
typedef float v8f __attribute__((ext_vector_type(8)));

// ---------------------------------------------------------------------------
// K1: wt[b,t] = sigmoid(dot(x[b,t,:], w) + bias)  via V_WMMA_F32_16X16X4_F32.
// One wave handles 16 consecutive rows (t values). A = 16x4 tile of x,
// B = 4x16 with w broadcast across all 16 columns -> every column of D is the
// dot product; accumulate over K = 512 in steps of 4.
// A-layout (ISA 16x4 f32): lanes 0-15 -> M=0..15 with K={0,1}; lanes 16-31 ->
// same M with K={2,3}. B symmetric: lanes' K rows get w[k0+koff..+1].
// D-layout: lane 0 VGPR v = D[v][0] (rows 0..7), lane 16 VGPR v = D[v+8][0].
// ---------------------------------------------------------------------------
__global__ __launch_bounds__(256) void k1_gate(const float* __restrict__ x,
                                               const float* __restrict__ w,
                                               const float* __restrict__ bptr,
                                               float* __restrict__ wt) {
  __shared__ float sw[FF];
  const int tid = threadIdx.x;
  for (int i = tid; i < FF; i += 256) sw[i] = w[i];
  __syncthreads();

  const int wave = tid >> 5;
  const int lane = tid & 31;
  const long row0 = ((long)blockIdx.x * 8 + wave) * 16;   // 16 t-rows per wave
  const int m = lane & 15;
  const int koff = (lane >> 4) << 1;                      // 0 or 2
  const float* xrow = x + (row0 + m) * FF + koff;

  v8f c = {0.f, 0.f, 0.f, 0.f, 0.f, 0.f, 0.f, 0.f};
#pragma unroll 4
  for (int k0 = 0; k0 < FF; k0 += 4) {
    v2f a = *(const v2f*)(xrow + k0);                     // A 16x4 fragment
    v2f bf = *(const v2f*)(&sw[k0 + koff]);               // B: w replicated cols
    c = __builtin_amdgcn_wmma_f32_16x16x4_f32(
        /*neg_a=*/false, a, /*neg_b=*/false, bf,
        /*c_mod=*/(short)0, c, /*reuse_a=*/false, /*reuse_b=*/false);
  }

  const float bias = bptr[0];
  if (lane == 0 || lane == 16) {
    const long base = row0 + ((lane >> 4) << 3);
#pragma unroll
    for (int v = 0; v < 8; ++v) {
      float z = c[v] + bias;
      wt[base + v] = 1.0f / (1.0f + __expf(-z));
    }
  }
}

// ---------------------------------------------------------------------------
// K2: per-batch strict-local-minimum detection + left-packed compaction.
// starts = {0} U valleys (value t); ends = valleys (value min(t+2,T)) U {T-1
// slot -> value T}. One exclusive scan over valley flags gives both packings.
// ---------------------------------------------------------------------------
__global__ __launch_bounds__(256) void k2_segments(const float* __restrict__ wt,
                                                   int* __restrict__ starts,
                                                   int* __restrict__ ends,
                                                   int* __restrict__ counts) {
  __shared__ float sw[TT];
  __shared__ int ssum[256];
  const int b = blockIdx.x;
  const int tid = threadIdx.x;
  const float* wb = wt + (long)b * TT;
  for (int i = tid; i < TT; i += 256) sw[i] = wb[i];
  __syncthreads();

  const int PER = TT / 256;  // 8
  int vflag[PER];
  int flags = 0;
#pragma unroll
  for (int j = 0; j < PER; ++j) {
    int t = tid * PER + j;
    float cc = sw[t];
    float before = (t == 0) ? 0.f : sw[t - 1];        // sigmoid>0 => t=0 never valley
    float after = (t == TT - 1) ? 0.f : sw[t + 1];    // => t=T-1 never valley
    int v = (cc < before && cc < after) ? 1 : 0;
    vflag[j] = v;
    flags += v;
  }
  ssum[tid] = flags;
  __syncthreads();
  // Hillis-Steele inclusive scan over per-thread sums
  for (int off = 1; off < 256; off <<= 1) {
    int v = (tid >= off) ? ssum[tid - off] : 0;
    __syncthreads();
    ssum[tid] += v;
    __syncthreads();
  }
  const int excl = ssum[tid] - flags;
  const int total = ssum[255];  // number of valleys in this batch

  int* sb = starts + (long)b * TT;
  int* eb = ends + (long)b * TT;
  int r = excl;
#pragma unroll
  for (int j = 0; j < PER; ++j) {
    if (vflag[j]) {
      int t = tid * PER + j;
      sb[1 + r] = t;                                  // start slot (after slot 0)
      eb[r] = (t + 2 > TT) ? TT : (t + 2);            // min(t+2, T)
      ++r;
    }
  }
  if (tid == 0) {
    sb[0] = 0;
    counts[b] = total + 1;
  }
  if (tid == 255) {
    eb[total] = TT;                                   // min(T-1+2, T) == T
  }
}

// ---------------------------------------------------------------------------
// K3: max_counts across batches, new_lengths, write mask (as 0/1 floats).
// ---------------------------------------------------------------------------
__global__ __launch_bounds__(256) void k3_mask(const int* __restrict__ counts,
                                               const int* __restrict__ seq_len,
                                               float* __restrict__ mask) {
  __shared__ int snl[BB];
  const int tid = threadIdx.x;
  if (tid == 0) {
    int mc = 0;
    for (int b = 0; b < BB; ++b) mc = (counts[b] > mc) ? counts[b] : mc;
    int len0 = seq_len[0] < TT ? seq_len[0] : TT;
    for (int b = 0; b < BB; ++b) {
      int len = seq_len[b] < TT ? seq_len[b] : TT;
      snl[b] = (int)((float)len / (float)len0 * (float)mc);  // trunc, matches jnp cast
    }
  }
  __syncthreads();
  for (int i = tid; i < BB * TT; i += 256) {
    int b = i / TT;
    int t = i - b * TT;
    mask[i] = (t < snl[b]) ? 1.0f : 0.0f;
  }
}

// ---------------------------------------------------------------------------
// K4: pooled[b,i,:] = sum_{t in [s,e)} wt[t]*x[b,t,:] / max(sum wt, 1e-6).
// One 128-thread block per (segment slot, batch); each thread owns 4 channels
// -> fully coalesced float4 row streaming. Slots >= counts[b] written as 0.
// ---------------------------------------------------------------------------
__global__ __launch_bounds__(128) void k4_pool(const float* __restrict__ x,
                                               const float* __restrict__ wt,
                                               const int* __restrict__ starts,
                                               const int* __restrict__ ends,
                                               const int* __restrict__ counts,
                                               float* __restrict__ pooled) {
  const int b = blockIdx.y;
  const int i = blockIdx.x;
  const int tid = threadIdx.x;
  float* out = pooled + ((long)b * TT + i) * FF + tid * 4;

  if (i >= counts[b]) {
    *(float4*)out = make_float4(0.f, 0.f, 0.f, 0.f);
    return;
  }
  const int s = starts[(long)b * TT + i];
  const int e = ends[(long)b * TT + i];
  const float* wb = wt + (long)b * TT;
  const float* xb = x + (long)b * TT * FF + tid * 4;

  float a0 = 0.f, a1 = 0.f, a2 = 0.f, a3 = 0.f, den = 0.f;
  for (int t = s; t < e; ++t) {
    float wv = wb[t];
    const float* xr = xb + (long)t * FF;
    __builtin_prefetch(xr + FF, 0, 1);                // global_prefetch_b8 next row
    float4 xv = *(const float4*)xr;
    a0 += wv * xv.x;
    a1 += wv * xv.y;
    a2 += wv * xv.z;
    a3 += wv * xv.w;
    den += wv;
  }
  den = (den < 1e-6f) ? 1e-6f : den;
  *(float4*)out = make_float4(a0 / den, a1 / den, a2 / den, a3 / den);
}

// ---------------------------------------------------------------------------
extern "C" void kernel_launch(void* const* d_in, const int* in_sizes, int n_in,
                              void* d_out, int out_size, void* d_ws, size_t ws_size,
                              hipStream_t stream) {
  const float* x = (const float*)d_in[0];     // [B,T,F]
  const float* w = (const float*)d_in[1];     // [F,1]
  const float* bias = (const float*)d_in[2];  // [1]
  const int* seq = (const int*)d_in[3];       // [B]

  float* pooled = (float*)d_out;                       // [B,T,F]
  float* mask = pooled + (long)BB * TT * FF;           // [B,T] as 0/1 floats

  char* ws = (char*)d_ws;
  float* wt = (float*)ws;                               // B*T floats
  int* starts = (int*)(ws + (size_t)BB * TT * 4);       // B*T ints
  int* ends = (int*)(ws + (size_t)2 * BB * TT * 4);     // B*T ints
  int* counts = (int*)(ws + (size_t)3 * BB * TT * 4);   // B ints

  // K1: 32768 rows / 16 per wave / 8 waves per block = 256 blocks
  k1_gate<<<(BB * TT / 16) / 8, 256, 0, stream>>>(x, w, bias, wt);
  // K2: one block per batch
  k2_segments<<<BB, 256, 0, stream>>>(wt, starts, ends, counts);
  // K3: single block
  k3_mask<<<1, 256, 0, stream>>>(counts, seq, mask);
  // K4: one block per (segment slot, batch)
  dim3 g4(TT, BB);
  k4_pool<<<g4, 128, 0, stream>>>(x, wt, starts, ends, counts, pooled);
}